// RepNorm1d_79748952752439
// MI455X (gfx1250) — compile-verified
//
#include <hip/hip_runtime.h>
#include <hip/hip_bf16.h>

#define N_ROWS   50000
#define D_DIM    1024
#define C_DIM    256
#define NORM_EPS 0.0001f
#define SELF_EPS 0.01f

#define TILES_PER_BLOCK 5
#define ROWS_PER_BLOCK  80      // 5 tiles * 16 rows (WMMA M)
#define NBLOCKS         625     // 50000 / 80
#define X0_PITCH        272     // 256 + 16 pad: bank-conflict-free WMMA gather

typedef float        v2f __attribute__((ext_vector_type(2)));
typedef float        v8f __attribute__((ext_vector_type(8)));
typedef unsigned int v4u __attribute__((ext_vector_type(4)));
typedef int          v8i __attribute__((ext_vector_type(8)));
typedef int          v4i __attribute__((ext_vector_type(4)));

// ---- Tensor Data Mover: DMA `nrows` contiguous rows of f32 (row = D_DIM elems)
// from global into LDS at byte offset lds_bytes. Issued once per executing wave.
__device__ __forceinline__ void tdm_load_rows(const float* gsrc,
                                              unsigned lds_bytes, int nrows) {
  unsigned long long ga = (unsigned long long)(uintptr_t)gsrc;
  v4u g0;
  g0[0] = 1u;                                         // count = 1 valid D#
  g0[1] = lds_bytes;                                  // lds_addr [63:32]
  g0[2] = (unsigned)ga;                               // global_addr[31:0]
  g0[3] = (unsigned)((ga >> 32) & 0x1FFFFFFull)       // global_addr[56:32]
          | (2u << 30);                               // type = 2 ("image")
  const unsigned td0 = D_DIM, td1 = 1u << 20;         // tensor dims (no OOB hit)
  const unsigned tile0 = D_DIM, ts0 = D_DIM;          // tile width, dim0 stride
  v8i g1;
  g1[0] = (int)(2u << 16);                            // data_size = 4B
  g1[1] = (int)((td0 & 0xFFFFu) << 16);               // tensor_dim0 lo
  g1[2] = (int)(((td0 >> 16) & 0xFFFFu) | ((td1 & 0xFFFFu) << 16));
  g1[3] = (int)(((td1 >> 16) & 0xFFFFu) | ((tile0 & 0xFFFFu) << 16));
  g1[4] = (int)(nrows & 0xFFFF);                      // tile_dim1; tile_dim2=0
  g1[5] = (int)ts0;                                   // tensor_dim0_stride lo
  g1[6] = 0;
  g1[7] = 0;
  v4i z4 = {0, 0, 0, 0};
#if defined(__clang_major__) && (__clang_major__ >= 23)
  v8i z8 = {0, 0, 0, 0, 0, 0, 0, 0};
  __builtin_amdgcn_tensor_load_to_lds(g0, g1, z4, z4, z8, 0);
#else
  __builtin_amdgcn_tensor_load_to_lds(g0, g1, z4, z4, 0);
#endif
}

// Pass 1: per 16-row tile -> segmented sum-of-squares, x0, divisor, x2,
// and WMMA column-reduction of x0 / x0^2 into per-block channel partials.
__global__ __launch_bounds__(256) void repnorm_pass1(
    const float* __restrict__ x, const float* __restrict__ beta,
    const int* __restrict__ rep, float* __restrict__ out_x1,
    float* __restrict__ out_x2, float* __restrict__ pS1,
    float* __restrict__ pS2)
{
  __shared__ float s_x[16 * D_DIM];            // 64 KB x tile (TDM target)
  __shared__ float s_x0[16 * X0_PITCH];        // x0 tile
  __shared__ float s_div[16 * X0_PITCH];       // divisor tile
  __shared__ int   s_rep[D_DIM];
  __shared__ int   s_bnd[C_DIM + 1];
  __shared__ float s_beta[C_DIM];

  const int tid  = threadIdx.x;
  const int wave = tid >> 5;
  const int lane = tid & 31;
  const int b    = blockIdx.x;

  for (int i = tid; i < D_DIM; i += 256) s_rep[i] = rep[i];
  s_beta[tid] = beta[tid];
  __syncthreads();
  { // channel c covers columns [bnd[c], bnd[c+1]) since rep_index is sorted
    int lo = 0, hi = D_DIM;
    const int c = tid;
    while (lo < hi) { int mid = (lo + hi) >> 1; if (s_rep[mid] < c) lo = mid + 1; else hi = mid; }
    s_bnd[c] = lo;
    if (tid == 0) s_bnd[C_DIM] = D_DIM;
  }

  const v2f ones = {1.0f, 1.0f};               // all-ones 16x4 A-matrix
  v8f accS1[2], accS2[2];
  #pragma unroll
  for (int k = 0; k < 2; ++k) {
    accS1[k] = (v8f){0, 0, 0, 0, 0, 0, 0, 0};
    accS2[k] = (v8f){0, 0, 0, 0, 0, 0, 0, 0};
  }

  const unsigned lds_x = (unsigned)(uintptr_t)(void*)&s_x[0];

  for (int t = 0; t < TILES_PER_BLOCK; ++t) {
    const int row0 = b * ROWS_PER_BLOCK + t * 16;
    __syncthreads();                           // tile buffers free for reuse
    // each wave DMAs its 2 rows (8 KB) into LDS via the Tensor Data Mover
    tdm_load_rows(x + (size_t)(row0 + wave * 2) * D_DIM,
                  lds_x + (unsigned)(wave * 2 * D_DIM * 4), 2);
    __builtin_amdgcn_s_wait_tensorcnt(0);
    __syncthreads();                           // full 16-row tile visible

    // Phase 1: thread c owns channel c; segmented sum over its run, 16 rows.
    {
      const int   c  = tid;
      const int   d0 = s_bnd[c], d1 = s_bnd[c + 1];
      const float bc = s_beta[c];
      #pragma unroll
      for (int r = 0; r < 16; ++r) {
        const float* xr = &s_x[r * D_DIM];
        float s = 0.0f;
        for (int d = d0; d < d1; ++d) { float v = xr[d]; s = fmaf(v, v, s); }
        const float x0 = sqrtf(s + NORM_EPS * NORM_EPS);
        s_x0[r * X0_PITCH + c]  = x0;
        s_div[r * X0_PITCH + c] = fabsf(x0 + bc) + SELF_EPS;
        out_x1[(size_t)(row0 + r) * C_DIM + c] = x0;   // staged; pass 3 BN-normalizes
      }
    }
    __syncthreads();

    // Phase 2: x2 = x / divisor[rep_index], coalesced stores.
    #pragma unroll
    for (int r = 0; r < 16; ++r) {
      const float* xr   = &s_x[r * D_DIM];
      float*       orow = out_x2 + (size_t)(row0 + r) * D_DIM;
      #pragma unroll
      for (int j = 0; j < 4; ++j) {
        const int d = j * 256 + tid;
        orow[d] = xr[d] / s_div[r * X0_PITCH + s_rep[d]];
      }
    }

    // Phase 3: WMMA column-sums of x0 and x0^2 over the 16 rows.
    // D = ones(16x4) x B(4x16) + C  => every output row holds the column sums.
    {
      const int half = lane >> 4;              // B layout: lanes 0-15 K 0-1, 16-31 K 2-3
      const int n    = lane & 15;
      #pragma unroll
      for (int k = 0; k < 2; ++k) {            // each wave owns 2 channel groups
        const int cc = (wave * 2 + k) * 16 + n;
        v8f a1 = accS1[k], a2 = accS2[k];
        #pragma unroll
        for (int kk = 0; kk < 4; ++kk) {       // 4 K-steps cover rows 0..15
          const int r0 = kk * 4 + half * 2;
          v2f bv;
          bv.x = s_x0[r0 * X0_PITCH + cc];           // K = 0 / 2
          bv.y = s_x0[(r0 + 1) * X0_PITCH + cc];     // K = 1 / 3
          a1 = __builtin_amdgcn_wmma_f32_16x16x4_f32(false, ones, false, bv,
                                                     (short)0, a1, false, false);
          v2f b2; b2.x = bv.x * bv.x; b2.y = bv.y * bv.y;
          a2 = __builtin_amdgcn_wmma_f32_16x16x4_f32(false, ones, false, b2,
                                                     (short)0, a2, false, false);
        }
        accS1[k] = a1; accS2[k] = a2;
      }
    }
  }

  // Partials: all M rows of D are identical (A was all ones); row M=0 lives in
  // accumulator VGPR0 on lanes 0..15.
  if (lane < 16) {
    #pragma unroll
    for (int k = 0; k < 2; ++k) {
      const int c = (wave * 2 + k) * 16 + lane;
      pS1[(size_t)b * C_DIM + c] = accS1[k][0];
      pS2[(size_t)b * C_DIM + c] = accS2[k][0];
    }
  }
}

// Pass 2: deterministic fp64 reduction of block partials -> mean, 1/std.
__global__ __launch_bounds__(256) void repnorm_pass2(
    const float* __restrict__ pS1, const float* __restrict__ pS2,
    float* __restrict__ mean_out, float* __restrict__ istd_out)
{
  const int c = threadIdx.x;
  double s1 = 0.0, s2 = 0.0;
  for (int b = 0; b < NBLOCKS; ++b) {
    s1 += (double)pS1[(size_t)b * C_DIM + c];
    s2 += (double)pS2[(size_t)b * C_DIM + c];
  }
  const double mean = s1 / (double)N_ROWS;
  double var = s2 / (double)N_ROWS - mean * mean;
  if (var < 0.0) var = 0.0;
  mean_out[c] = (float)mean;
  istd_out[c] = (float)(1.0 / sqrt(var + 1e-5));
}

// Pass 3: in-place BN on the staged x0 -> x1.
__global__ __launch_bounds__(256) void repnorm_pass3(
    float* __restrict__ out_x1, const float* __restrict__ mean,
    const float* __restrict__ istd, int rows_per_block)
{
  const int   c = threadIdx.x;
  const float m = mean[c];
  const float s = istd[c];
  int r0 = blockIdx.x * rows_per_block;
  int r1 = r0 + rows_per_block;
  if (r1 > N_ROWS) r1 = N_ROWS;
  for (int r = r0; r < r1; ++r) {
    const size_t i = (size_t)r * C_DIM + c;
    out_x1[i] = (out_x1[i] - m) * s;
  }
}

extern "C" void kernel_launch(void* const* d_in, const int* in_sizes, int n_in,
                              void* d_out, int out_size, void* d_ws, size_t ws_size,
                              hipStream_t stream) {
  const float* x    = (const float*)d_in[0];
  const float* beta = (const float*)d_in[1];
  const int*   rep  = (const int*)d_in[2];

  float* out_x1 = (float*)d_out;                          // N x C
  float* out_x2 = out_x1 + (size_t)N_ROWS * C_DIM;        // N x D

  float* pS1  = (float*)d_ws;                             // NBLOCKS x C
  float* pS2  = pS1 + (size_t)NBLOCKS * C_DIM;            // NBLOCKS x C
  float* mean = pS2 + (size_t)NBLOCKS * C_DIM;            // C
  float* istd = mean + C_DIM;                             // C

  repnorm_pass1<<<NBLOCKS, 256, 0, stream>>>(x, beta, rep, out_x1, out_x2, pS1, pS2);
  repnorm_pass2<<<1, 256, 0, stream>>>(pS1, pS2, mean, istd);
  repnorm_pass3<<<(N_ROWS + 31) / 32, 256, 0, stream>>>(out_x1, mean, istd, 32);
}